// ReportDecoder_16260746183149
// MI455X (gfx1250) — compile-verified
//
#include <hip/hip_runtime.h>
#include <math.h>
#include <stdint.h>

// ReportDecoder (LSTM) for MI455X / gfx1250, fp32 WMMA path + async-LDS DMA.
//
// Sizes: B=256, HW=196, E=1024, H=1024, L=128, T=48, FC_OUT=1154.
// Strategy:
//   - img_mean once (HBM-bound, 205MB read).
//   - Hoist img_mean @ W_ih[:, :E]^T + b_ih out of the time loop (g_img).
//   - Per step: gates = g_img + b_hh + h@W_hh^T + lab_t@W_ihL^T  (WMMA GEMM),
//               pointwise LSTM cell, then fused FC + heads + length mask.
//   - GEMM K-loop double-buffers tiles with GLOBAL_LOAD_ASYNC_TO_LDS_B128
//     (ASYNCcnt), overlapping L2->LDS traffic with v_wmma compute. This is
//     the key latency optimization: the recurrence serializes 48 steps, so
//     per-tile memory latency is on the critical path.
// All GEMMs use V_WMMA_F32_16X16X4_F32 (fp32 matches reference numerics;
// workload is latency/L2-bound so low-precision WMMA buys nothing).
//
// Workspace usage: (3*256*1024 + 2*256*4096) floats = ~11.5 MB of d_ws.

typedef __attribute__((ext_vector_type(2))) float v2f;
typedef __attribute__((ext_vector_type(4))) float f4;
typedef __attribute__((ext_vector_type(8))) float v8f;

#define BM 128
#define BN 64
#define BK 32
#define LDSK 36  // BK + 4 pad: 16B-aligned rows (144B), conflict-free frag reads

__device__ __forceinline__ v8f wmma4(v2f a, v2f b, v8f c) {
  // (neg_a, A, neg_b, B, c_mod, C, reuse_a, reuse_b)
  return __builtin_amdgcn_wmma_f32_16x16x4_f32(false, a, false, b, (short)0, c,
                                               false, false);
}

__device__ __forceinline__ float sigmoidf_(float x) {
  return 1.0f / (1.0f + __expf(-x));
}

// Async direct-to-LDS 128-bit load (VGLOBAL encoding, tracked by ASYNCcnt).
// lds_off: byte offset within the workgroup's LDS allocation (generic-pointer
// low 32 bits); g: 16B-aligned global address.
__device__ __forceinline__ void async_b128(unsigned lds_off, const float* g) {
  asm volatile("global_load_async_to_lds_b128 %0, %1, off"
               :: "v"(lds_off), "v"(g)
               : "memory");
}

// Accumulate acc += A_blk(BM x K) * W_blk(BN x K)^T.
// A pre-offset to the block's M rows (row-major, stride lda).
// W pre-offset to the block's N rows (row-major, stride ldw); rows >= Nrem are
// clamped to row Nrem-1 (their products land in discarded cols >= N).
// K must be a multiple of BK; Nrem >= 1. All 256 threads participate.
// Alds/Blds are double buffers: [2][BM*LDSK] / [2][BN*LDSK].
__device__ __forceinline__ void gemm_accum(const float* __restrict__ A, int lda,
                                           const float* __restrict__ W, int ldw,
                                           int K, int Nrem,
                                           float* Alds, float* Blds,
                                           v8f acc[4]) {
  const int tid = threadIdx.x;
  const int lane = tid & 31;
  const int wave = tid >> 5;
  const int mrow = wave * 16 + (lane & 15);   // wave's M row within block tile
  const int kb = (lane >> 4) << 1;            // lanes 0-15 -> K {0,1}, 16-31 -> {2,3}
  const int nk = K / BK;

  // Issue one BK tile (A: 4 x b128, W: 2 x b128 per thread; exactly 6 async
  // instruction issues per wave -> ASYNCcnt bookkeeping below is exact).
  auto issue_tile = [&](int kt, int buf) {
    const int k0 = kt * BK;
    float* Ab = Alds + buf * (BM * LDSK);
    float* Bb = Blds + buf * (BN * LDSK);
#pragma unroll
    for (int i = 0; i < 4; ++i) {
      int pos = tid + 256 * i;
      int r = pos >> 3;
      int c4 = (pos & 7) << 2;
      async_b128((unsigned)(uintptr_t)(Ab + r * LDSK + c4),
                 A + (size_t)r * lda + k0 + c4);
    }
#pragma unroll
    for (int i = 0; i < 2; ++i) {
      int pos = tid + 256 * i;
      int r = pos >> 3;
      int c4 = (pos & 7) << 2;
      int rr = (r < Nrem) ? r : (Nrem - 1);  // clamp: keeps issue count uniform
      async_b128((unsigned)(uintptr_t)(Bb + r * LDSK + c4),
                 W + (size_t)rr * ldw + k0 + c4);
    }
  };

  issue_tile(0, 0);
  if (nk > 1) issue_tile(1, 1);

  for (int kt = 0; kt < nk; ++kt) {
    // Async loads complete in order: <=6 outstanding means tile kt landed.
    if (kt + 1 < nk)
      asm volatile("s_wait_asynccnt 0x6" ::: "memory");
    else
      asm volatile("s_wait_asynccnt 0x0" ::: "memory");
    __syncthreads();  // all waves' DMA contributions visible

    const float* Ab = Alds + (kt & 1) * (BM * LDSK);
    const float* Bb = Blds + (kt & 1) * (BN * LDSK);
#pragma unroll
    for (int kk = 0; kk < BK; kk += 4) {
      // A frag (16x4, ISA 7.12.2): lane%16 = M, VGPR v holds K = kb+v
      v2f a;
      a.x = Ab[mrow * LDSK + kk + kb];
      a.y = Ab[mrow * LDSK + kk + kb + 1];
#pragma unroll
      for (int j = 0; j < 4; ++j) {
        // B frag (4x16): lane%16 = N, VGPR v holds K = kb+v; B[k][n] = W[n][k]
        int n = j * 16 + (lane & 15);
        v2f b;
        b.x = Bb[n * LDSK + kk + kb];
        b.y = Bb[n * LDSK + kk + kb + 1];
        acc[j] = wmma4(a, b, acc[j]);
      }
    }
    __syncthreads();  // all waves done reading buf (kt&1) before DMA reuse
    if (kt + 2 < nk) issue_tile(kt + 2, kt & 1);
  }
}

// D/C 16x16 layout: VGPR v -> row (v + 8*(lane>=16)), col = lane%16.
#define EPILOGUE_ROWCOL(row, col)                                  \
  int row = m0 + wave * 16 + v + ((lane >> 4) << 3);               \
  int col = n0 + j * 16 + (lane & 15);

// ---------------- img_mean ----------------
__global__ void mean_kernel(const float* __restrict__ image,
                            float* __restrict__ img_mean) {
  int e = blockIdx.x * 256 + threadIdx.x;  // 0..1023
  int b = blockIdx.y;
  const float* p = image + (size_t)b * 196 * 1024 + e;
  float s = 0.f;
#pragma unroll 4
  for (int hw = 0; hw < 196; ++hw) s += p[(size_t)hw * 1024];
  img_mean[(size_t)b * 1024 + e] = s * (1.0f / 196.0f);
}

// ---------------- generic GEMM + bias (+tanh) ----------------
template <int ACT>  // 0: none, 1: tanh
__global__ void gemm_bias_act(const float* __restrict__ A, int lda,
                              const float* __restrict__ W, int ldw,
                              const float* __restrict__ bias,
                              float* __restrict__ out, int ldo, int N, int K) {
  __shared__ float Alds[2 * BM * LDSK];
  __shared__ float Blds[2 * BN * LDSK];
  const int n0 = blockIdx.x * BN;
  const int m0 = blockIdx.y * BM;
  v8f acc[4] = {};
  gemm_accum(A + (size_t)m0 * lda, lda, W + (size_t)n0 * ldw, ldw, K, N - n0,
             Alds, Blds, acc);
  const int lane = threadIdx.x & 31, wave = threadIdx.x >> 5;
#pragma unroll
  for (int j = 0; j < 4; ++j)
#pragma unroll
    for (int v = 0; v < 8; ++v) {
      EPILOGUE_ROWCOL(row, col)
      if (col < N) {
        float x = acc[j][v] + bias[col];
        if (ACT == 1) x = tanhf(x);
        out[(size_t)row * ldo + col] = x;
      }
    }
}

// ---------------- per-step gates GEMM ----------------
// gates = (g_img + b_hh) + h @ W_hh^T + lab_t @ W_ih[:,E:]^T
__global__ void gates_kernel(const float* __restrict__ h,
                             const float* __restrict__ W_hh,
                             const float* __restrict__ lab, int lda_lab,
                             const float* __restrict__ W_ihL,
                             const float* __restrict__ gimg,
                             const float* __restrict__ b_hh,
                             float* __restrict__ gates) {
  __shared__ float Alds[2 * BM * LDSK];
  __shared__ float Blds[2 * BN * LDSK];
  const int n0 = blockIdx.x * BN;
  const int m0 = blockIdx.y * BM;
  const int lane = threadIdx.x & 31, wave = threadIdx.x >> 5;
  v8f acc[4];
#pragma unroll
  for (int j = 0; j < 4; ++j)
#pragma unroll
    for (int v = 0; v < 8; ++v) {
      EPILOGUE_ROWCOL(row, col)
      acc[j][v] = gimg[(size_t)row * 4096 + col] + b_hh[col];
    }
  gemm_accum(h + (size_t)m0 * 1024, 1024, W_hh + (size_t)n0 * 1024, 1024, 1024,
             4096 - n0, Alds, Blds, acc);
  gemm_accum(lab + (size_t)m0 * lda_lab, lda_lab, W_ihL + (size_t)n0 * 1152,
             1152, 128, 4096 - n0, Alds, Blds, acc);
#pragma unroll
  for (int j = 0; j < 4; ++j)
#pragma unroll
    for (int v = 0; v < 8; ++v) {
      EPILOGUE_ROWCOL(row, col)
      gates[(size_t)row * 4096 + col] = acc[j][v];
    }
}

// ---------------- pointwise LSTM cell ----------------
__global__ void cell_kernel(const float* __restrict__ gates,
                            float* __restrict__ h, float* __restrict__ m) {
  int idx = blockIdx.x * blockDim.x + threadIdx.x;  // 0 .. 256*1024-1
  int b = idx >> 10, j = idx & 1023;
  const float* g = gates + (size_t)b * 4096;
  float gi = g[j], gf = g[1024 + j], gg = g[2048 + j], go = g[3072 + j];
  float mm = sigmoidf_(gf) * m[idx] + sigmoidf_(gi) * tanhf(gg);
  m[idx] = mm;
  h[idx] = sigmoidf_(go) * tanhf(mm);
}

// ---------------- FC + heads + mask ----------------
__global__ void fc_kernel(const float* __restrict__ h,
                          const float* __restrict__ W_fc,
                          const float* __restrict__ b_fc,
                          const int* __restrict__ length, int t,
                          float* __restrict__ out_lab,
                          float* __restrict__ out_top,
                          float* __restrict__ out_tmp,
                          float* __restrict__ out_stp) {
  __shared__ float Alds[2 * BM * LDSK];
  __shared__ float Blds[2 * BN * LDSK];
  const int n0 = blockIdx.x * BN;
  const int m0 = blockIdx.y * BM;
  v8f acc[4] = {};
  gemm_accum(h + (size_t)m0 * 1024, 1024, W_fc + (size_t)n0 * 1024, 1024, 1024,
             1154 - n0, Alds, Blds, acc);
  const int lane = threadIdx.x & 31, wave = threadIdx.x >> 5;
#pragma unroll
  for (int j = 0; j < 4; ++j)
#pragma unroll
    for (int v = 0; v < 8; ++v) {
      EPILOGUE_ROWCOL(row, col)
      if (col < 1154) {
        float x = fmaxf(acc[j][v] + b_fc[col], 0.0f);  // relu
        float mask = (t < length[row]) ? 1.0f : 0.0f;
        size_t bt = (size_t)row * 48 + t;
        if (col < 128)
          out_lab[bt * 128 + col] = x * mask;
        else if (col < 1152)
          out_top[bt * 1024 + (col - 128)] = x * mask;
        else if (col == 1152)
          out_tmp[bt] = __expf(x) * mask;
        else
          out_stp[bt] = sigmoidf_(x) * mask;
      }
    }
}

extern "C" void kernel_launch(void* const* d_in, const int* in_sizes, int n_in,
                              void* d_out, int out_size, void* d_ws,
                              size_t ws_size, hipStream_t stream) {
  const float* image  = (const float*)d_in[0];   // (256,196,1024)
  const float* label  = (const float*)d_in[1];   // (256,48,128)
  const int*   length = (const int*)d_in[2];     // (256,)
  const float* W_h  = (const float*)d_in[3];     // (1024,1024)
  const float* b_h  = (const float*)d_in[4];
  const float* W_m  = (const float*)d_in[5];     // (1024,1024)
  const float* b_m  = (const float*)d_in[6];
  const float* W_ih = (const float*)d_in[7];     // (4096,1152)
  const float* W_hh = (const float*)d_in[8];     // (4096,1024)
  const float* b_ih = (const float*)d_in[9];
  const float* b_hh = (const float*)d_in[10];
  const float* W_fc = (const float*)d_in[11];    // (1154,1024)
  const float* b_fc = (const float*)d_in[12];

  float* out = (float*)d_out;
  float* out_lab = out;                                // (256,48,128)
  float* out_top = out_lab + (size_t)256 * 48 * 128;   // (256,48,1024)
  float* out_tmp = out_top + (size_t)256 * 48 * 1024;  // (256,48,1)
  float* out_stp = out_tmp + (size_t)256 * 48;         // (256,48,1)

  float* ws = (float*)d_ws;
  float* img_mean = ws;  ws += (size_t)256 * 1024;
  float* hbuf     = ws;  ws += (size_t)256 * 1024;
  float* mbuf     = ws;  ws += (size_t)256 * 1024;
  float* gimg     = ws;  ws += (size_t)256 * 4096;
  float* gates    = ws;  ws += (size_t)256 * 4096;

  // 1) img_mean (HBM-bound)
  mean_kernel<<<dim3(4, 256), 256, 0, stream>>>(image, img_mean);

  // 2) h0 = tanh(img_mean @ W_h^T + b_h), m0 = tanh(img_mean @ W_m^T + b_m)
  gemm_bias_act<1><<<dim3(16, 2), 256, 0, stream>>>(img_mean, 1024, W_h, 1024,
                                                    b_h, hbuf, 1024, 1024, 1024);
  gemm_bias_act<1><<<dim3(16, 2), 256, 0, stream>>>(img_mean, 1024, W_m, 1024,
                                                    b_m, mbuf, 1024, 1024, 1024);
  // 3) g_img = img_mean @ W_ih[:, :1024]^T + b_ih  (hoisted out of the scan)
  gemm_bias_act<0><<<dim3(64, 2), 256, 0, stream>>>(img_mean, 1024, W_ih, 1152,
                                                    b_ih, gimg, 4096, 4096, 1024);

  // 4) recurrence (weights stay L2-resident across steps)
  for (int t = 0; t < 48; ++t) {
    gates_kernel<<<dim3(64, 2), 256, 0, stream>>>(
        hbuf, W_hh, label + (size_t)t * 128, 48 * 128, W_ih + 1024, gimg, b_hh,
        gates);
    cell_kernel<<<dim3(1024), 256, 0, stream>>>(gates, hbuf, mbuf);
    fc_kernel<<<dim3(19, 2), 256, 0, stream>>>(hbuf, W_fc, b_fc, length, t,
                                               out_lab, out_top, out_tmp,
                                               out_stp);
  }
}